// CartBondedWholePoseScoringModule_23081154249117
// MI455X (gfx1250) — compile-verified
//
#include <hip/hip_runtime.h>
#include <stdint.h>

// Problem constants (from reference)
#define P_POSES 32
#define A_ATOMS 4096
#define S_TOTAL 1500000
#define T_HASH  1048576u
#define EPS_F   1e-8f

#define BLOCK 256
#define ITER  4

struct f3 { float x, y, z; };
__device__ __forceinline__ f3 sub3(f3 a, f3 b) { return {a.x - b.x, a.y - b.y, a.z - b.z}; }
__device__ __forceinline__ float dot3(f3 a, f3 b) { return a.x * b.x + a.y * b.y + a.z * b.z; }
__device__ __forceinline__ f3 cross3(f3 a, f3 b) {
  return {a.y * b.z - a.z * b.y, a.z * b.x - a.x * b.z, a.x * b.y - a.y * b.x};
}

// ---- CDNA5 async global->LDS (ASYNCcnt-tracked) --------------------------
__device__ __forceinline__ void async_ld_b128(uint32_t lds_off, const void* g) {
  asm volatile("global_load_async_to_lds_b128 %0, %1, off"
               :: "v"(lds_off), "v"(g) : "memory");
}
__device__ __forceinline__ void async_ld_b32(uint32_t lds_off, const void* g) {
  asm volatile("global_load_async_to_lds_b32 %0, %1, off"
               :: "v"(lds_off), "v"(g) : "memory");
}
__device__ __forceinline__ void wait_async0() {
#if __has_builtin(__builtin_amdgcn_s_wait_asynccnt)
  __builtin_amdgcn_s_wait_asynccnt(0);
#else
  asm volatile("s_wait_asynccnt 0x0" ::: "memory");
#endif
}
__device__ __forceinline__ void wait_ds0() {
#if __has_builtin(__builtin_amdgcn_s_wait_dscnt)
  __builtin_amdgcn_s_wait_dscnt(0);
#else
  asm volatile("s_wait_dscnt 0x0" ::: "memory");
#endif
}
// --------------------------------------------------------------------------

__global__ void zero_out_kernel(float* out) {
  if (threadIdx.x < P_POSES) out[threadIdx.x] = 0.0f;
}

__global__ void __launch_bounds__(BLOCK)
cartbonded_kernel(const float* __restrict__ coords,   // (P, A, 3)
                  const float* __restrict__ hashv,    // (T, 3)
                  const int*  __restrict__ atoms,     // (S, 4)
                  const int*  __restrict__ pose,      // (S,)
                  const int*  __restrict__ uids,      // (P, A)
                  float* __restrict__ out)            // (P,)
{
  __shared__ int4  sAtoms[2][BLOCK];
  __shared__ int   sPose[2][BLOCK];
  __shared__ float sAcc[P_POSES];

  const int tid = threadIdx.x;
  if (tid < P_POSES) sAcc[tid] = 0.0f;
  __syncthreads();

  const int4* atoms4 = (const int4*)atoms;

  auto issue = [&](int buf, int chunk) {
    int s = chunk * BLOCK + tid;                 // fits in 31 bits
    if (s > S_TOTAL - 1) s = S_TOTAL - 1;        // keep addresses in-bounds
    async_ld_b128((uint32_t)(size_t)&sAtoms[buf][tid], atoms4 + s);
    async_ld_b32 ((uint32_t)(size_t)&sPose [buf][tid], pose + s);
  };

  const int chunk0 = (int)blockIdx.x * ITER;

  // prime the pipeline
  issue(0, chunk0);

  for (int i = 0; i < ITER; ++i) {
    wait_async0();                               // current buffer resident in LDS

    const int s  = (chunk0 + i) * BLOCK + tid;
    const int4 at = sAtoms[i & 1][tid];
    const int  pp = sPose [i & 1][tid];

    // Ensure the DS reads above have completed before the async engine may
    // overwrite this LDS region again (async LDS writes are unordered vs
    // in-flight DS reads), then prefetch the next tile into the other buffer.
    wait_ds0();
    if (i + 1 < ITER) issue((i + 1) & 1, chunk0 + i + 1);

    if (s < S_TOTAL) {
      const bool v0 = at.x >= 0, v1 = at.y >= 0, v2 = at.z >= 0, v3 = at.w >= 0;
      const int i0 = v0 ? at.x : 0;
      const int i1 = v1 ? at.y : 0;
      const int i2 = v2 ? at.z : 0;
      const int i3 = v3 ? at.w : 0;

      const int* __restrict__ urow = uids + pp * A_ATOMS;
      const unsigned usum = (v0 ? (unsigned)urow[i0] : 0u) +
                            (v1 ? (unsigned)urow[i1] : 0u) +
                            (v2 ? (unsigned)urow[i2] : 0u) +
                            (v3 ? (unsigned)urow[i3] : 0u);
      const unsigned key = usum & (T_HASH - 1u);  // T is 2^20

      const float* __restrict__ hp = hashv + 3u * key;
      const float K = hp[0], x0 = hp[1], period = hp[2];

      const float* __restrict__ cb = coords + (unsigned)pp * (A_ATOMS * 3);
      const f3 p0 = {cb[3 * i0], cb[3 * i0 + 1], cb[3 * i0 + 2]};
      const f3 p1 = {cb[3 * i1], cb[3 * i1 + 1], cb[3 * i1 + 2]};
      const f3 p2 = {cb[3 * i2], cb[3 * i2 + 1], cb[3 * i2 + 2]};
      const f3 p3 = {cb[3 * i3], cb[3 * i3 + 1], cb[3 * i3 + 2]};

      const int nv = (int)v0 + (int)v1 + (int)v2 + (int)v3;

      float e;
      if (nv == 2) {
        const f3 d10 = sub3(p1, p0);
        const float d = sqrtf(dot3(d10, d10) + EPS_F);
        const float t = d - x0;
        e = K * t * t;
      } else if (nv == 3) {
        const f3 u = sub3(p0, p1), w = sub3(p2, p1);
        float c = dot3(u, w) / (sqrtf(dot3(u, u) + EPS_F) * sqrtf(dot3(w, w) + EPS_F));
        c = fminf(0.999999f, fmaxf(-0.999999f, c));
        const float t = acosf(c) - x0;
        e = K * t * t;
      } else {
        const f3 b1 = sub3(p1, p0), b2 = sub3(p2, p1), b3 = sub3(p3, p2);
        const f3 n1 = cross3(b1, b2), n2 = cross3(b2, b3);
        const float inv = 1.0f / sqrtf(dot3(b2, b2) + EPS_F);
        const f3 b2n = {b2.x * inv, b2.y * inv, b2.z * inv};
        const f3 m1 = cross3(n1, b2n);
        const float phi = atan2f(dot3(m1, n2), dot3(n1, n2) + EPS_F);
        e = K * (1.0f + cosf(period * phi - x0));
      }

      atomicAdd(&sAcc[pp], e);  // ds_add_f32 (LDS float atomic)
    }
  }

  __syncthreads();
  if (tid < P_POSES) {
#if defined(__HIP_DEVICE_COMPILE__)
    unsafeAtomicAdd(&out[tid], sAcc[tid]);  // native global_atomic_add_f32
#else
    atomicAdd(&out[tid], sAcc[tid]);
#endif
  }
}

extern "C" void kernel_launch(void* const* d_in, const int* in_sizes, int n_in,
                              void* d_out, int out_size, void* d_ws, size_t ws_size,
                              hipStream_t stream) {
  const float* coords = (const float*)d_in[0];   // (32, 4096, 3) f32
  const float* hashv  = (const float*)d_in[1];   // (T, 3) f32
  const int*   atoms  = (const int*)d_in[2];     // (S, 4) i32
  const int*   pose   = (const int*)d_in[3];     // (S,) i32
  const int*   uids   = (const int*)d_in[4];     // (32, 4096) i32
  float* out = (float*)d_out;                    // (32,) f32

  zero_out_kernel<<<1, 32, 0, stream>>>(out);

  const int chunks = (S_TOTAL + BLOCK - 1) / BLOCK;   // 5860
  const int grid   = (chunks + ITER - 1) / ITER;      // 1465
  cartbonded_kernel<<<grid, BLOCK, 0, stream>>>(coords, hashv, atoms, pose, uids, out);
}